// MultiScaleSelfAttention_19232863551877
// MI455X (gfx1250) — compile-verified
//
#include <hip/hip_runtime.h>
#include <hip/hip_bf16.h>
#include <math.h>

#define Bn 4
#define Cc 256
#define Hh 56
#define Ww 56
#define Nn 3136
#define EDc 128
#define HD 32
#define SCALE 0.17677669529663687f  /* 32^-0.5 */

typedef __attribute__((ext_vector_type(2))) float v2f;
typedef __attribute__((ext_vector_type(8))) float v8f;

__device__ __forceinline__ v8f wmma_f32(v2f a, v2f b, v8f c) {
  return __builtin_amdgcn_wmma_f32_16x16x4_f32(false, a, false, b, (short)0, c, false, false);
}

// ---------------------------------------------------------------------------
// LayerNorm over channels, per (b, pixel)
// ---------------------------------------------------------------------------
__global__ void ln_kernel(const float* __restrict__ x, const float* __restrict__ g,
                          const float* __restrict__ b, float* __restrict__ xln) {
  int idx = blockIdx.x * blockDim.x + threadIdx.x;
  if (idx >= Bn * Nn) return;
  int bb = idx / Nn, p = idx - bb * Nn;
  const float* xp = x + (size_t)bb * Cc * Nn + p;
  float s = 0.f, s2 = 0.f;
  for (int c = 0; c < Cc; c++) { float v = xp[(size_t)c * Nn]; s += v; s2 += v * v; }
  float mean = s * (1.f / Cc);
  float var  = s2 * (1.f / Cc) - mean * mean;
  float inv  = rsqrtf(var + 1e-5f);
  float* op = xln + (size_t)bb * Cc * Nn + p;
  for (int c = 0; c < Cc; c++)
    op[(size_t)c * Nn] = (xp[(size_t)c * Nn] - mean) * inv * g[c] + b[c];
}

// ---------------------------------------------------------------------------
// 1x1 conv as WMMA f32 GEMM. One wave per 16x16 (out-ch x pixel) tile.
// ---------------------------------------------------------------------------
template <int IC>
__global__ void __launch_bounds__(32)
conv1x1_wmma_kernel(const float* __restrict__ xin, const float* __restrict__ add,
                    const float* __restrict__ w, const float* __restrict__ bias,
                    const float* __restrict__ sc, const float* __restrict__ sh,
                    int relu, float* __restrict__ y, int OC) {
  const int lane = threadIdx.x;
  const int hf = lane >> 4;
  const int l16 = lane & 15;
  const int p0 = blockIdx.x * 16;
  const int o0 = blockIdx.y * 16;
  const int bb = blockIdx.z;
  const size_t xbase = (size_t)bb * IC * Nn;
  v8f acc = {};
#pragma unroll 8
  for (int s = 0; s < IC / 4; s++) {
    v2f a;
    a.x = w[(size_t)(o0 + l16) * IC + s * 4 + 2 * hf + 0];
    a.y = w[(size_t)(o0 + l16) * IC + s * 4 + 2 * hf + 1];
    v2f b;
    size_t cx = xbase + (size_t)(s * 4 + 2 * hf) * Nn + p0 + l16;
    b.x = xin[cx];
    b.y = xin[cx + Nn];
    if (add) { b.x += add[cx]; b.y += add[cx + Nn]; }
    acc = wmma_f32(a, b, acc);
  }
#pragma unroll
  for (int r = 0; r < 8; r++) {
    int oc = o0 + r + 8 * hf;
    float val = acc[r];
    if (bias) val += bias[oc];
    if (sc) val = val * sc[oc] + sh[oc];
    if (relu) val = fmaxf(val, 0.f);
    y[((size_t)bb * OC + oc) * Nn + p0 + l16] = val;
  }
}

// ---------------------------------------------------------------------------
// Depthwise 3x3 (pad 1) + scale + bias + relu
// ---------------------------------------------------------------------------
__global__ void dw3_kernel(const float* __restrict__ xin, const float* __restrict__ dw,
                           const float* __restrict__ s1, const float* __restrict__ b1,
                           float* __restrict__ y) {
  int idx = blockIdx.x * blockDim.x + threadIdx.x;
  if (idx >= Bn * EDc * Nn) return;
  int p = idx % Nn;
  int ch = (idx / Nn) % EDc;
  int bb = idx / (Nn * EDc);
  int px = p % Ww, py = p / Ww;
  const float* base = xin + ((size_t)bb * EDc + ch) * Nn;
  const float* w9 = dw + ch * 9;
  float acc = 0.f;
#pragma unroll
  for (int dy = -1; dy <= 1; dy++) {
    int yy = py + dy;
    if (yy < 0 || yy >= Hh) continue;
#pragma unroll
    for (int dx = -1; dx <= 1; dx++) {
      int xx = px + dx;
      if (xx < 0 || xx >= Ww) continue;
      acc = fmaf(base[yy * Ww + xx], w9[(dy + 1) * 3 + (dx + 1)], acc);
    }
  }
  float r = acc * s1[ch] + b1[ch];
  y[((size_t)bb * EDc + ch) * Nn + p] = fmaxf(r, 0.f);
}

// ---------------------------------------------------------------------------
// moe_pool7: per (b, head2, d, 7, 7) avg & max over 8x8 block
// ---------------------------------------------------------------------------
__global__ void pool_stats_kernel(const float* __restrict__ t, float* __restrict__ avg,
                                  float* __restrict__ mx) {
  int idx = blockIdx.x * blockDim.x + threadIdx.x;
  if (idx >= Bn * 2 * HD * 49) return;
  int cell = idx % 49;
  int px = cell % 7, py = cell / 7;
  int d  = (idx / 49) % HD;
  int h2 = (idx / (49 * HD)) % 2;
  int bb = idx / (49 * HD * 2);
  const float* base = t + ((size_t)bb * EDc + 64 + h2 * HD + d) * Nn;
  float s = 0.f, m = -__builtin_inff();
  for (int iy = 0; iy < 8; iy++)
    for (int ix = 0; ix < 8; ix++) {
      float v = base[(py * 8 + iy) * Ww + (px * 8 + ix)];
      s += v;
      m = fmaxf(m, v);
    }
  avg[idx] = s * (1.f / 64.f);
  mx[idx] = m;
}

__global__ void pool_gate_kernel(const float* __restrict__ avg, const float* __restrict__ mx,
                                 float* __restrict__ og) {
  __shared__ float red[256];
  int base = blockIdx.x * HD * 49;  // blockIdx.x = bb*2 + h2
  int tid = threadIdx.x;
  float s = 0.f;
  for (int i = tid; i < HD * 49; i += 256) {
    float d = avg[base + i] - mx[base + i];
    s += d * d;
  }
  red[tid] = s;
  __syncthreads();
  for (int st = 128; st > 0; st >>= 1) {
    if (tid < st) red[tid] += red[tid + st];
    __syncthreads();
  }
  float msq = red[0] * (1.f / (HD * 49));
  float gate = 1.f / (1.f + __expf(msq));  // sigmoid(-msq)
  for (int i = tid; i < HD * 49; i += 256)
    og[base + i] = gate * mx[base + i] + (1.f - gate) * avg[base + i];
}

// ---------------------------------------------------------------------------
// Dense attention (heads 0,1), flash-style, WMMA f32.
// 128-thread block = 4 waves = 4 query tiles sharing K/V tiles staged in LDS
// (4x less L2 traffic). Softmax denominators accumulated via an extra WMMA
// (P @ ones); only the row max needs cross-lane shuffles.
// ---------------------------------------------------------------------------
__global__ void __launch_bounds__(128) attn_a_kernel(const float* __restrict__ q,
                                                     const float* __restrict__ k,
                                                     const float* __restrict__ v,
                                                     float* __restrict__ ao) {
  __shared__ float ksh[32 * 72];     // K tile [ch][key], stride 72: QK frags conflict-free
  __shared__ float vsh[64 * 36];     // V tile transposed [key][d], stride 36
  __shared__ float pls[4][16 * 65];  // per-wave P staging (same-wave LDS is in-order)

  const int tid = threadIdx.x;
  const int wid = tid >> 5;
  const int lane = tid & 31;
  const int hf = lane >> 4;
  const int l16 = lane & 15;
  const int qbase = (blockIdx.x * 4 + wid) * 16;
  const int h = blockIdx.y;   // head 0 or 1
  const int bb = blockIdx.z;
  const size_t ch0 = ((size_t)bb * EDc + h * HD) * Nn;
  float* pw = pls[wid];

  // Q fragments (A layout, 8 k-steps of 4): A[m=l16, kd=s*4+2*hf+j]
  v2f aq[8];
#pragma unroll
  for (int s = 0; s < 8; s++) {
    aq[s].x = q[ch0 + (size_t)(s * 4 + 2 * hf + 0) * Nn + qbase + l16];
    aq[s].y = q[ch0 + (size_t)(s * 4 + 2 * hf + 1) * Nn + qbase + l16];
  }

  v8f o0 = {}, o1 = {}, o2 = {};   // o2 = running softmax denominators (replicated)
  float mi[8];
#pragma unroll
  for (int r = 0; r < 8; r++) mi[r] = -__builtin_inff();
  const v2f ones = {1.f, 1.f};

  for (int kt = 0; kt < Nn / 64; kt++) {
    const int kbase = kt * 64;
    __syncthreads();  // previous K/V tile fully consumed by all waves
    // Stage K tile: 32 ch x 64 keys, coalesced float4 per thread
#pragma unroll
    for (int pass = 0; pass < 4; pass++) {
      int ch = pass * 8 + (tid >> 4);
      int c4 = (tid & 15) * 4;
      float4 val = *(const float4*)(k + ch0 + (size_t)ch * Nn + kbase + c4);
      *(float4*)&ksh[ch * 72 + c4] = val;
    }
    // Stage V tile transposed: vsh[key][d] (coalesced reads, conflict-free stores)
#pragma unroll
    for (int i = 0; i < 16; i++) {
      int idx = i * 128 + tid;
      int d = idx >> 6, key = idx & 63;
      vsh[key * 36 + d] = v[ch0 + (size_t)d * Nn + kbase + key];
    }
    __syncthreads();

    // S tiles for 64 keys
    v8f sa[4];
#pragma unroll
    for (int t = 0; t < 4; t++) {
      sa[t] = (v8f){};
#pragma unroll
      for (int s = 0; s < 8; s++) {
        v2f bk;
        bk.x = ksh[(s * 4 + 2 * hf + 0) * 72 + t * 16 + l16];
        bk.y = ksh[(s * 4 + 2 * hf + 1) * 72 + t * 16 + l16];
        sa[t] = wmma_f32(aq[s], bk, sa[t]);
      }
    }
    // scale + online softmax over the 64-key batch
#pragma unroll
    for (int r = 0; r < 8; r++) {
#pragma unroll
      for (int t = 0; t < 4; t++) sa[t][r] *= SCALE;
      float rm = fmaxf(fmaxf(sa[0][r], sa[1][r]), fmaxf(sa[2][r], sa[3][r]));
#pragma unroll
      for (int off = 8; off >= 1; off >>= 1) rm = fmaxf(rm, __shfl_xor(rm, off, 32));
      float mnew = fmaxf(mi[r], rm);
      float corr = __expf(mi[r] - mnew);
      mi[r] = mnew;
      o0[r] *= corr;
      o1[r] *= corr;
      o2[r] *= corr;
#pragma unroll
      for (int t = 0; t < 4; t++) sa[t][r] = __expf(sa[t][r] - mnew);
    }
    // Stage P (C layout) -> per-wave LDS -> A layout (no barrier needed: LDS ops
    // of one wave complete in order)
#pragma unroll
    for (int t = 0; t < 4; t++)
#pragma unroll
      for (int r = 0; r < 8; r++)
        pw[(r + 8 * hf) * 65 + t * 16 + l16] = sa[t][r];
    // O += P @ V^T ; denom += P @ ones
#pragma unroll
    for (int s2 = 0; s2 < 16; s2++) {
      v2f ap;
      ap.x = pw[l16 * 65 + s2 * 4 + 2 * hf + 0];
      ap.y = pw[l16 * 65 + s2 * 4 + 2 * hf + 1];
      int krow = (s2 * 4 + 2 * hf) * 36;
      v2f bv0, bv1;
      bv0.x = vsh[krow + l16];
      bv0.y = vsh[krow + 36 + l16];
      bv1.x = vsh[krow + 16 + l16];
      bv1.y = vsh[krow + 36 + 16 + l16];
      o0 = wmma_f32(ap, bv0, o0);
      o1 = wmma_f32(ap, bv1, o1);
      o2 = wmma_f32(ap, ones, o2);
    }
  }
  // Normalize and write: element r -> row q = r+8*hf, col d = l16 (/+16)
#pragma unroll
  for (int r = 0; r < 8; r++) {
    float inv = 1.f / o2[r];
    ao[ch0 + (size_t)(l16) * Nn + qbase + r + 8 * hf] = o0[r] * inv;
    ao[ch0 + (size_t)(16 + l16) * Nn + qbase + r + 8 * hf] = o1[r] * inv;
  }
}

// ---------------------------------------------------------------------------
// Pooled attention (heads 2,3): 49 keys, one thread per query pixel.
// ---------------------------------------------------------------------------
__global__ void attn_b_kernel(const float* __restrict__ q, const float* __restrict__ kg,
                              const float* __restrict__ vg, float* __restrict__ ao) {
  int idx = blockIdx.x * blockDim.x + threadIdx.x;
  if (idx >= Bn * 2 * Nn) return;
  int p = idx % Nn;
  int h2 = (idx / Nn) % 2;
  int bb = idx / (2 * Nn);
  size_t ch0 = ((size_t)bb * EDc + 64 + h2 * HD) * Nn;
  float qv[HD];
#pragma unroll
  for (int d = 0; d < HD; d++) qv[d] = q[ch0 + (size_t)d * Nn + p];
  const float* kp = kg + (size_t)(bb * 2 + h2) * HD * 49;
  const float* vp = vg + (size_t)(bb * 2 + h2) * HD * 49;
  float lg[49];
  float mx = -__builtin_inff();
  for (int m = 0; m < 49; m++) {
    float s = 0.f;
#pragma unroll
    for (int d = 0; d < HD; d++) s = fmaf(qv[d], kp[d * 49 + m], s);
    s *= SCALE;
    lg[m] = s;
    mx = fmaxf(mx, s);
  }
  float sum = 0.f;
  for (int m = 0; m < 49; m++) { float e = __expf(lg[m] - mx); lg[m] = e; sum += e; }
  float inv = 1.f / sum;
  for (int d = 0; d < HD; d++) {
    float acc = 0.f;
    for (int m = 0; m < 49; m++) acc = fmaf(lg[m], vp[d * 49 + m], acc);
    ao[ch0 + (size_t)d * Nn + p] = acc * inv;
  }
}

// ---------------------------------------------------------------------------
// Bilinear attention maps.
// ---------------------------------------------------------------------------
__device__ __forceinline__ void bil_coord(int i, int isz, int osz, int& lo, int& hi, float& f) {
  float c = ((float)i + 0.5f) * ((float)isz / (float)osz) - 0.5f;
  c = fminf(fmaxf(c, 0.f), (float)(isz - 1));
  float fl = floorf(c);
  lo = (int)fl;
  hi = min(lo + 1, isz - 1);
  f = c - fl;
}

__global__ void maps_a_kernel(const float* __restrict__ q, const float* __restrict__ k,
                              float* __restrict__ map) {
  __shared__ float qs[2][HD];
  __shared__ float red[128];
  __shared__ float stat[4];  // mx0, mx1, sum0, sum1
  int tid = threadIdx.x;
  int i = blockIdx.x;
  int h = blockIdx.y & 1;
  int bb = blockIdx.y >> 1;
  size_t ch0 = ((size_t)bb * EDc + h * HD) * Nn;
  int y[2]; float fy;
  bil_coord(i, Nn, 100, y[0], y[1], fy);
  if (tid < 2 * HD) {
    int rr = tid / HD, d = tid % HD;
    qs[rr][d] = q[ch0 + (size_t)d * Nn + y[rr]];
  }
  __syncthreads();
  float lmx[2] = {-__builtin_inff(), -__builtin_inff()};
  for (int j = tid; j < Nn; j += 128) {
    float s0 = 0.f, s1 = 0.f;
    for (int d = 0; d < HD; d++) {
      float kv = k[ch0 + (size_t)d * Nn + j];
      s0 = fmaf(qs[0][d], kv, s0);
      s1 = fmaf(qs[1][d], kv, s1);
    }
    lmx[0] = fmaxf(lmx[0], s0 * SCALE);
    lmx[1] = fmaxf(lmx[1], s1 * SCALE);
  }
  for (int rr = 0; rr < 2; rr++) {
    red[tid] = lmx[rr];
    __syncthreads();
    for (int st = 64; st > 0; st >>= 1) {
      if (tid < st) red[tid] = fmaxf(red[tid], red[tid + st]);
      __syncthreads();
    }
    if (tid == 0) stat[rr] = red[0];
    __syncthreads();
  }
  float m0 = stat[0], m1 = stat[1];
  float lsum[2] = {0.f, 0.f};
  for (int j = tid; j < Nn; j += 128) {
    float s0 = 0.f, s1 = 0.f;
    for (int d = 0; d < HD; d++) {
      float kv = k[ch0 + (size_t)d * Nn + j];
      s0 = fmaf(qs[0][d], kv, s0);
      s1 = fmaf(qs[1][d], kv, s1);
    }
    lsum[0] += __expf(s0 * SCALE - m0);
    lsum[1] += __expf(s1 * SCALE - m1);
  }
  for (int rr = 0; rr < 2; rr++) {
    red[tid] = lsum[rr];
    __syncthreads();
    for (int st = 64; st > 0; st >>= 1) {
      if (tid < st) red[tid] += red[tid + st];
      __syncthreads();
    }
    if (tid == 0) stat[2 + rr] = red[0];
    __syncthreads();
  }
  float inv0 = 1.f / stat[2], inv1 = 1.f / stat[3];
  if (tid < 100) {
    int x0, x1; float fx;
    bil_coord(tid, Nn, 100, x0, x1, fx);
    int xs[2] = {x0, x1};
    float a[4];
    for (int rr = 0; rr < 2; rr++)
      for (int cc = 0; cc < 2; cc++) {
        float s = 0.f;
        for (int d = 0; d < HD; d++)
          s = fmaf(qs[rr][d], k[ch0 + (size_t)d * Nn + xs[cc]], s);
        s *= SCALE;
        a[rr * 2 + cc] = __expf(s - (rr ? m1 : m0)) * (rr ? inv1 : inv0);
      }
    float top = a[0] * (1.f - fx) + a[1] * fx;
    float bot = a[2] * (1.f - fx) + a[3] * fx;
    map[((size_t)h * Bn + bb) * 10000 + i * 100 + tid] = top * (1.f - fy) + bot * fy;
  }
}

__global__ void maps_b_kernel(const float* __restrict__ q, const float* __restrict__ kg,
                              float* __restrict__ map) {
  __shared__ float lg[2][49];
  __shared__ float stat[4];
  int tid = threadIdx.x;
  int i = blockIdx.x;
  int h2 = blockIdx.y & 1;
  int bb = blockIdx.y >> 1;
  size_t ch0 = ((size_t)bb * EDc + 64 + h2 * HD) * Nn;
  const float* kp = kg + (size_t)(bb * 2 + h2) * HD * 49;
  int y[2]; float fy;
  bil_coord(i, Nn, 100, y[0], y[1], fy);
  if (tid < 98) {
    int rr = tid / 49, m = tid % 49;
    float s = 0.f;
    for (int d = 0; d < HD; d++)
      s = fmaf(q[ch0 + (size_t)d * Nn + y[rr]], kp[d * 49 + m], s);
    lg[rr][m] = s * SCALE;
  }
  __syncthreads();
  if (tid < 2) {
    float mx = -__builtin_inff();
    for (int m = 0; m < 49; m++) mx = fmaxf(mx, lg[tid][m]);
    float sm = 0.f;
    for (int m = 0; m < 49; m++) sm += __expf(lg[tid][m] - mx);
    stat[tid] = mx;
    stat[2 + tid] = sm;
  }
  __syncthreads();
  if (tid < 100) {
    int x0, x1; float fx;
    bil_coord(tid, 49, 100, x0, x1, fx);
    float a00 = __expf(lg[0][x0] - stat[0]) / stat[2];
    float a01 = __expf(lg[0][x1] - stat[0]) / stat[2];
    float a10 = __expf(lg[1][x0] - stat[1]) / stat[3];
    float a11 = __expf(lg[1][x1] - stat[1]) / stat[3];
    float top = a00 * (1.f - fx) + a01 * fx;
    float bot = a10 * (1.f - fx) + a11 * fx;
    map[((size_t)(2 + h2) * Bn + bb) * 10000 + i * 100 + tid] = top * (1.f - fy) + bot * fy;
  }
}

// ---------------------------------------------------------------------------
extern "C" void kernel_launch(void* const* d_in, const int* in_sizes, int n_in,
                              void* d_out, int out_size, void* d_ws, size_t ws_size,
                              hipStream_t stream) {
  (void)in_sizes; (void)n_in; (void)out_size; (void)ws_size;
  const float* x    = (const float*)d_in[0];
  const float* ln_g = (const float*)d_in[1];
  const float* ln_b = (const float*)d_in[2];
  const float *W[3], *Bi[3], *DW[3], *S1[3], *B1[3], *PW[3], *S2[3], *B2[3];
  for (int p = 0; p < 3; p++) {
    const int o = 3 + p * 8;
    W[p]  = (const float*)d_in[o + 0];
    Bi[p] = (const float*)d_in[o + 1];
    DW[p] = (const float*)d_in[o + 2];
    S1[p] = (const float*)d_in[o + 3];
    B1[p] = (const float*)d_in[o + 4];
    PW[p] = (const float*)d_in[o + 5];
    S2[p] = (const float*)d_in[o + 6];
    B2[p] = (const float*)d_in[o + 7];
  }
  const float* ow = (const float*)d_in[27];
  const float* ob = (const float*)d_in[28];

  float* out = (float*)d_out;
  float* map = out + (size_t)Bn * Cc * Nn;

  float* ws = (float*)d_ws;
  size_t off = 0;
  float* xln = ws + off;  off += (size_t)Bn * Cc * Nn;
  float* tA  = ws + off;  off += (size_t)Bn * EDc * Nn;
  float* tB  = ws + off;  off += (size_t)Bn * EDc * Nn;
  float* qkv[3];
  for (int p = 0; p < 3; p++) { qkv[p] = ws + off; off += (size_t)Bn * EDc * Nn; }
  float* aout = ws + off; off += (size_t)Bn * EDc * Nn;
  float* avgb = ws + off; off += (size_t)Bn * 2 * HD * 49;
  float* mxb  = ws + off; off += (size_t)Bn * 2 * HD * 49;
  float* kgb  = ws + off; off += (size_t)Bn * 2 * HD * 49;
  float* vgb  = ws + off; off += (size_t)Bn * 2 * HD * 49;

  // 1) LayerNorm
  ln_kernel<<<(Bn * Nn + 255) / 256, 256, 0, stream>>>(x, ln_g, ln_b, xln);

  // 2) q/k/v projection pipelines (1x1 convs on the WMMA pipe)
  dim3 cgE(Nn / 16, EDc / 16, Bn);
  int dwth = Bn * EDc * Nn;
  for (int p = 0; p < 3; p++) {
    conv1x1_wmma_kernel<Cc><<<cgE, 32, 0, stream>>>(xln, nullptr, W[p], Bi[p], nullptr, nullptr, 0, tA, EDc);
    dw3_kernel<<<(dwth + 255) / 256, 256, 0, stream>>>(tA, DW[p], S1[p], B1[p], tB);
    conv1x1_wmma_kernel<EDc><<<cgE, 32, 0, stream>>>(tB, nullptr, PW[p], nullptr, S2[p], B2[p], 1, qkv[p], EDc);
  }

  // 3) moe_pool7 for k and v (heads 2,3)
  int pst = Bn * 2 * HD * 49;
  pool_stats_kernel<<<(pst + 255) / 256, 256, 0, stream>>>(qkv[1], avgb, mxb);
  pool_gate_kernel<<<8, 256, 0, stream>>>(avgb, mxb, kgb);
  pool_stats_kernel<<<(pst + 255) / 256, 256, 0, stream>>>(qkv[2], avgb, mxb);
  pool_gate_kernel<<<8, 256, 0, stream>>>(avgb, mxb, vgb);

  // 4) attention (4 query tiles per workgroup share K/V tiles in LDS)
  dim3 ag(Nn / 64, 2, Bn);
  attn_a_kernel<<<ag, 128, 0, stream>>>(qkv[0], qkv[1], qkv[2], aout);
  attn_b_kernel<<<(Bn * 2 * Nn + 127) / 128, 128, 0, stream>>>(qkv[0], kgb, vgb, aout);

  // 5) residual + output projection (WMMA GEMM with fused residual add)
  dim3 cgC(Nn / 16, Cc / 16, Bn);
  conv1x1_wmma_kernel<EDc><<<cgC, 32, 0, stream>>>(aout, qkv[2], ow, ob, nullptr, nullptr, 0, out, Cc);

  // 6) attention maps
  dim3 mg(100, 8);
  maps_a_kernel<<<mg, 128, 0, stream>>>(qkv[0], qkv[1], map);
  maps_b_kernel<<<mg, 128, 0, stream>>>(qkv[0], kgb, map);
}